// DecoupledMoEContainer_59751585022466
// MI455X (gfx1250) — compile-verified
//
#include <hip/hip_runtime.h>

typedef float v2f __attribute__((ext_vector_type(2)));
typedef float v4f __attribute__((ext_vector_type(4)));
typedef float v8f __attribute__((ext_vector_type(8)));

#define NB       128          // batch
#define CIN      64
#define COUT     64
#define HW       3136         // 56*56
#define PXC      64           // pixels per block
#define NCHUNK   (HW / PXC)   // 49
#define KPAIRS   (CIN / 2)    // 32 interleaved channel pairs in LDS
#define KP_S     160          // kpair row stride in floats: 160 mod 64(dwords) == 32
                              // -> half-wave 0 uses banks 0..31, half-wave 1 banks 32..63

// One block = one (sample, 64-pixel chunk). 8 waves of 32.
// Wave w: M-tile (w&3) -> output channels [(w&3)*16, +16); N-tiles at (w>>2)*32 and +16.
// LDS layout: xs[kpair*KP_S + px*2 + (ch&1)]  => each B fragment is ONE ds_load_b64.
__launch_bounds__(256, 2)
__global__ void moe_1x1_wmma_f32(const float* __restrict__ x,
                                 const float* __restrict__ weights,
                                 const int*   __restrict__ indices,
                                 const float* __restrict__ Ws,   // [COUT, CIN]
                                 const float* __restrict__ bs,   // [COUT]
                                 const float* __restrict__ Wr,   // [7, COUT, CIN]
                                 const float* __restrict__ br,   // [7, COUT]
                                 float* __restrict__ out) {
    __shared__ float xs[KPAIRS * KP_S];   // 20 KB

    const int blk   = blockIdx.x;
    const int b     = blk / NCHUNK;
    const int chunk = blk - b * NCHUNK;
    const int p0    = chunk * PXC;

    const int   t   = threadIdx.x;
    const float wgt = weights[b];
    const int   idx = indices[b];

    const float* xb = x + (size_t)b * (CIN * HW) + p0;

    // ---- Stage x tile into LDS, K-pair interleaved --------------------------
    // Per iter: 256 threads cover 16 kpairs x 64 px. Thread t: kpair = t>>4,
    // px4 = (t&15)*4. Two coalesced b128 row loads, four interleaved b64 stores.
#pragma unroll
    for (int it = 0; it < 2; ++it) {
        const int kp  = it * 16 + (t >> 4);
        const int px4 = (t & 15) << 2;
        v4f r0 = *(const v4f*)(xb + (size_t)(2 * kp)     * HW + px4);
        v4f r1 = *(const v4f*)(xb + (size_t)(2 * kp + 1) * HW + px4);
        float* d = xs + kp * KP_S + px4 * 2;
#pragma unroll
        for (int i = 0; i < 4; ++i) {
            v2f p; p.x = r0[i]; p.y = r1[i];
            *(v2f*)(d + 2 * i) = p;
        }
    }
    __syncthreads();

    // ---- Wave / lane decomposition (wave32) ---------------------------------
    const int wave  = t >> 5;
    const int lane  = t & 31;
    const int m     = wave & 3;            // M-tile
    const int nbase = (wave >> 2) * 32;    // first of two N-tiles
    const int half  = lane >> 4;           // 0: K=k0,k0+1  1: K=k0+2,k0+3 (ISA A/B layout)
    const int lrow  = lane & 15;           // A: row M; B/D: column N

    // A-fragment source rows of W_shared / W_routed[idx] (W_eff built on the fly)
    const float* wsRow = Ws + (size_t)(m * 16 + lrow) * CIN + half * 2;
    const float* wrRow = Wr + ((size_t)idx * COUT + m * 16 + lrow) * CIN + half * 2;

    // B-fragment base in LDS: kpair = 2k + half, px = nbase + lrow
    const float* bbase = xs + half * KP_S + (nbase + lrow) * 2;

    v8f acc0 = {};
    v8f acc1 = {};

#pragma unroll
    for (int k = 0; k < 16; ++k) {       // K = 64 in steps of 4
        // A fragment: W_eff[m*16+lrow][4k + half*2 .. +1]  (one b64 load each)
        v2f aS = *(const v2f*)(wsRow + 4 * k);
        v2f aR = *(const v2f*)(wrRow + 4 * k);
        v2f a;
        a.x = __builtin_fmaf(wgt, aR.x, aS.x);
        a.y = __builtin_fmaf(wgt, aR.y, aS.y);

        // B fragments: single ds_load_b64 each, already in WMMA pair layout
        const float* bp = bbase + (2 * k) * KP_S;
        v2f b0 = *(const v2f*)(bp);        // N-tile at nbase
        v2f b1 = *(const v2f*)(bp + 32);   // N-tile at nbase+16

        acc0 = __builtin_amdgcn_wmma_f32_16x16x4_f32(false, a, false, b0,
                                                     (short)0, acc0, false, false);
        acc1 = __builtin_amdgcn_wmma_f32_16x16x4_f32(false, a, false, b1,
                                                     (short)0, acc1, false, false);
    }

    // ---- Store with fused bias (D layout: VGPR r -> M = r + half*8, N = lrow) ----
    float* ob = out + (size_t)b * (COUT * HW) + p0;
#pragma unroll
    for (int r = 0; r < 8; ++r) {
        const int row  = m * 16 + half * 8 + r;
        const float be = __builtin_fmaf(wgt, br[idx * COUT + row], bs[row]);
        const int col  = nbase + lrow;
        ob[(size_t)row * HW + col]      = acc0[r] + be;
        ob[(size_t)row * HW + col + 16] = acc1[r] + be;
    }
}

extern "C" void kernel_launch(void* const* d_in, const int* in_sizes, int n_in,
                              void* d_out, int out_size, void* d_ws, size_t ws_size,
                              hipStream_t stream) {
    const float* x   = (const float*)d_in[0];
    const float* w   = (const float*)d_in[1];
    const int*   idx = (const int*)  d_in[2];
    const float* Ws  = (const float*)d_in[3];
    const float* bsh = (const float*)d_in[4];
    const float* Wr  = (const float*)d_in[5];
    const float* br  = (const float*)d_in[6];
    float* out = (float*)d_out;

    dim3 grid(NB * NCHUNK);   // 6272 blocks
    dim3 block(256);          // 8 wave32
    moe_1x1_wmma_f32<<<grid, block, 0, stream>>>(x, w, idx, Ws, bsh, Wr, br, out);
}